// HANLayer_75333726371844
// MI455X (gfx1250) — compile-verified
//
#include <hip/hip_runtime.h>
#include <hip/hip_bf16.h>

// ---------------- problem constants (match reference) ----------------
#define NN   100000
#define EE   1600000
#define MM   3
#define HH   8
#define FF   16
#define DD   128           // HH*FF
#define INV  64            // in_size
#define HID  64
#define NEG_SLOPE 0.2f

typedef __attribute__((ext_vector_type(2))) float v2f;
typedef __attribute__((ext_vector_type(8))) float v8f;

// ------------- helpers: monotone uint encoding for float atomicMax -------------
__device__ __forceinline__ unsigned mono(float x) {
    unsigned b = __float_as_uint(x);
    return (b & 0x80000000u) ? ~b : (b | 0x80000000u);
}
__device__ __forceinline__ float demono(unsigned u) {
    return (u & 0x80000000u) ? __uint_as_float(u & 0x7FFFFFFFu)
                             : __uint_as_float(~u);
}

// ---------------- generic u32 fill (scalar, small buffers) ----------------
__global__ void fill_u32_kernel(unsigned* __restrict__ p, unsigned val, size_t n) {
    size_t i = (size_t)blockIdx.x * blockDim.x + threadIdx.x;
    if (i < n) p[i] = val;
}

// ---------------- b128 fill (large buffers; n4 = count of uint4) ----------------
__global__ void fill_u32x4_kernel(uint4* __restrict__ p, unsigned val, size_t n4) {
    size_t i = (size_t)blockIdx.x * blockDim.x + threadIdx.x;
    if (i < n4) p[i] = make_uint4(val, val, val, val);
}

// ---------------- feat = h @ W[m]  (f32 WMMA 16x16x4) ----------------
// grid.x = NN/16 node tiles, block = 128 threads = 4 waves.
// Wave w computes two 16x16 tiles at columns [32w, 32w+32).
__global__ __launch_bounds__(128)
void feat_wmma_kernel(const float* __restrict__ h,    // [NN, 64]
                      const float* __restrict__ Wm,   // [64, 128]
                      float* __restrict__ feat)       // [NN, 128]
{
    __shared__ float hs[16 * 65];   // 16x64 tile, padded to kill bank conflicts
    const int tid  = threadIdx.x;
    const int row0 = blockIdx.x * 16;

    // coalesced stage of the h tile into LDS
    for (int i = tid; i < 16 * 64; i += 128) {
        int r = i >> 6, c = i & 63;
        hs[r * 65 + c] = h[(size_t)(row0 + r) * INV + c];
    }
    __syncthreads();

    const int lane = tid & 31;
    const int wave = tid >> 5;
    const int mrow = lane & 15;     // M index
    const int kh   = lane >> 4;     // K half-select (0/1)
    const int n0   = wave * 32;     // this wave's first column tile

    v8f c0 = {};
    v8f c1 = {};

    for (int k0 = 0; k0 < INV; k0 += 4) {
        const int ka = k0 + 2 * kh;
        // A fragment (16x4 f32): lane l -> A[mrow][ka+v]
        v2f a;
        a.x = hs[mrow * 65 + ka];
        a.y = hs[mrow * 65 + ka + 1];
        // B fragments (4x16 f32): lane l -> B[ka+v][n]
        const float* Wp = Wm + (size_t)ka * DD;
        v2f b0, b1;
        b0.x = Wp[n0 + mrow];
        b0.y = Wp[DD + n0 + mrow];
        b1.x = Wp[n0 + 16 + mrow];
        b1.y = Wp[DD + n0 + 16 + mrow];

        c0 = __builtin_amdgcn_wmma_f32_16x16x4_f32(false, a, false, b0,
                                                   (short)0, c0, false, false);
        c1 = __builtin_amdgcn_wmma_f32_16x16x4_f32(false, a, false, b1,
                                                   (short)0, c1, false, false);
    }

    // D layout: VGPR g holds row (g + 8*kh), col = n0 + mrow
    for (int g = 0; g < 8; ++g) {
        const int r = row0 + g + 8 * kh;
        feat[(size_t)r * DD + n0 + mrow]      = c0[g];
        feat[(size_t)r * DD + n0 + 16 + mrow] = c1[g];
    }
}

// ---------------- el/er: per-(node, head) attention logits ----------------
__global__ void attn_lr_kernel(const float* __restrict__ feat,
                               const float* __restrict__ al,   // [H,F]
                               const float* __restrict__ ar,   // [H,F]
                               float* __restrict__ el,         // [NN,H]
                               float* __restrict__ er)         // [NN,H]
{
    int idx = blockIdx.x * blockDim.x + threadIdx.x;   // n*H + h
    if (idx >= NN * HH) return;
    const int hh = idx & 7;
    const float* f   = feat + (size_t)(idx >> 3) * DD + hh * FF;
    const float* alp = al + hh * FF;
    const float* arp = ar + hh * FF;
    float sl = 0.f, sr = 0.f;
#pragma unroll
    for (int i = 0; i < FF; ++i) {
        float v = f[i];
        sl += v * alp[i];
        sr += v * arp[i];
    }
    el[idx] = sl;
    er[idx] = sr;
}

// ---------------- edge pass 1: segment max over dst ----------------
__global__ void edge_max_kernel(const int* __restrict__ src,
                                const int* __restrict__ dst,
                                const float* __restrict__ el,
                                const float* __restrict__ er,
                                unsigned* __restrict__ mxu)
{
    int idx = blockIdx.x * blockDim.x + threadIdx.x;   // e*H + h
    if (idx >= EE * HH) return;
    const int e  = idx >> 3;
    const int hh = idx & 7;
    const int s  = src[e];
    const int d  = dst[e];
    float v = el[s * HH + hh] + er[d * HH + hh];
    v = (v > 0.f) ? v : NEG_SLOPE * v;
    atomicMax(&mxu[d * HH + hh], mono(v));
}

// ---------------- edge pass 2: segment sum of exp(e - mx) ----------------
__global__ void edge_sum_kernel(const int* __restrict__ src,
                                const int* __restrict__ dst,
                                const float* __restrict__ el,
                                const float* __restrict__ er,
                                const unsigned* __restrict__ mxu,
                                float* __restrict__ ssum)
{
    int idx = blockIdx.x * blockDim.x + threadIdx.x;   // e*H + h
    if (idx >= EE * HH) return;
    const int e  = idx >> 3;
    const int hh = idx & 7;
    const int s  = src[e];
    const int d  = dst[e];
    float v = el[s * HH + hh] + er[d * HH + hh];
    v = (v > 0.f) ? v : NEG_SLOPE * v;
    float a = expf(v - demono(mxu[d * HH + hh]));
    atomicAdd(&ssum[d * HH + hh], a);
}

// ---------------- edge pass 3: alpha-weighted scatter aggregate ----------------
// 32 threads per edge; each thread owns 4 contiguous features (one head slice):
// feat gathered as b128, alpha computed once per thread (4x less redundancy
// than thread-per-feature), 4 scalar f32 atomics into z.
__global__ __launch_bounds__(256)
void edge_agg_kernel(const int* __restrict__ src,
                     const int* __restrict__ dst,
                     const float* __restrict__ el,
                     const float* __restrict__ er,
                     const unsigned* __restrict__ mxu,
                     const float* __restrict__ ssum,
                     const float* __restrict__ feat,
                     float* __restrict__ z,          // [NN, MM, DD]
                     int mpath)
{
    long long idx = (long long)blockIdx.x * 256 + threadIdx.x;  // e*32 + lane
    if (idx >= (long long)EE * 32) return;
    const int e    = (int)(idx >> 5);
    const int lane = (int)(idx & 31);
    const int dd0  = lane * 4;        // feature base; all 4 within head dd0>>4
    const int hh   = dd0 >> 4;
    const int s    = src[e];
    const int d    = dst[e];

    float v = el[s * HH + hh] + er[d * HH + hh];
    v = (v > 0.f) ? v : NEG_SLOPE * v;
    const float a     = expf(v - demono(mxu[d * HH + hh]));
    const float alpha = a / (ssum[d * HH + hh] + 1e-9f);

    const float4 f4 = *(const float4*)(feat + (size_t)s * DD + dd0);
    float* zp = z + ((size_t)d * MM + mpath) * DD + dd0;
    atomicAdd(zp + 0, alpha * f4.x);
    atomicAdd(zp + 1, alpha * f4.y);
    atomicAdd(zp + 2, alpha * f4.z);
    atomicAdd(zp + 3, alpha * f4.w);
}

// ---------------- z += bias (broadcast over nodes), b128 ----------------
__global__ void bias_add_kernel(float4* __restrict__ z4, const float4* __restrict__ bias4) {
    size_t i = (size_t)blockIdx.x * blockDim.x + threadIdx.x;  // over float4s
    if (i >= (size_t)NN * MM * DD / 4) return;
    float4 zv = z4[i];
    float4 bv = bias4[i % (MM * DD / 4)];
    zv.x += bv.x; zv.y += bv.y; zv.z += bv.z; zv.w += bv.w;
    z4[i] = zv;
}

// ---------------- semantic attention: per-(n,m) score, mean over n ----------------
// block = 64 threads handles one (n, m); thread j owns hidden unit j.
__global__ __launch_bounds__(64)
void semantic_kernel(const float* __restrict__ z,
                     const float* __restrict__ w1,   // [DD, HID]
                     const float* __restrict__ b1,   // [HID]
                     const float* __restrict__ w2,   // [HID]
                     float* __restrict__ wsum)       // [MM]
{
    __shared__ float zsh[DD];
    __shared__ float red[HID];
    const int b = blockIdx.x;
    const int n = b / MM;
    const int m = b % MM;
    const int j = threadIdx.x;

    const float* zp = z + ((size_t)n * MM + m) * DD;
    zsh[j]      = zp[j];
    zsh[j + 64] = zp[j + 64];
    __syncthreads();

    float acc = b1[j];
    for (int d = 0; d < DD; ++d)
        acc += zsh[d] * w1[(size_t)d * HID + j];   // coalesced across j
    red[j] = tanhf(acc) * w2[j];
    __syncthreads();

    if (j == 0) {
        float t = 0.f;
#pragma unroll
        for (int k = 0; k < HID; ++k) t += red[k];
        atomicAdd(&wsum[m], t * (1.0f / (float)NN));
    }
}

// ---------------- beta = softmax over meta-paths ----------------
__global__ void beta_kernel(const float* __restrict__ wsum, float* __restrict__ beta) {
    float w0 = wsum[0], w1 = wsum[1], w2 = wsum[2];
    float mx = fmaxf(w0, fmaxf(w1, w2));
    float e0 = expf(w0 - mx), e1 = expf(w1 - mx), e2 = expf(w2 - mx);
    float s = e0 + e1 + e2;
    beta[0] = e0 / s;
    beta[1] = e1 / s;
    beta[2] = e2 / s;
}

// ---------------- out = sum_m beta[m] * z[:,m,:]  (b128) ----------------
__global__ void combine_kernel(const float4* __restrict__ z4,
                               const float* __restrict__ beta,
                               float4* __restrict__ out4)
{
    size_t i = (size_t)blockIdx.x * blockDim.x + threadIdx.x;  // over float4s
    if (i >= (size_t)NN * DD / 4) return;
    const size_t n  = i >> 5;                 // DD/4 = 32 float4s per node
    const int    q  = (int)(i & 31);
    const float b0 = beta[0], b1 = beta[1], b2 = beta[2];
    const float4* zp = z4 + n * (MM * DD / 4);
    const float4 a = zp[q];
    const float4 c = zp[DD / 4 + q];
    const float4 d = zp[2 * (DD / 4) + q];
    float4 o;
    o.x = b0 * a.x + b1 * c.x + b2 * d.x;
    o.y = b0 * a.y + b1 * c.y + b2 * d.y;
    o.z = b0 * a.z + b1 * c.z + b2 * d.z;
    o.w = b0 * a.w + b1 * c.w + b2 * d.w;
    out4[i] = o;
}

// =====================================================================
extern "C" void kernel_launch(void* const* d_in, const int* in_sizes, int n_in,
                              void* d_out, int out_size, void* d_ws, size_t ws_size,
                              hipStream_t stream) {
    const float* h      = (const float*)d_in[0];   // [NN, 64]
    const int*   edges  = (const int*)  d_in[1];   // [MM, 2, EE]
    const float* W      = (const float*)d_in[2];   // [MM, 64, 128]
    const float* attn_l = (const float*)d_in[3];   // [MM, H, F]
    const float* attn_r = (const float*)d_in[4];   // [MM, H, F]
    const float* bias   = (const float*)d_in[5];   // [MM, DD]
    const float* sa_w1  = (const float*)d_in[6];   // [DD, HID]
    const float* sa_b1  = (const float*)d_in[7];   // [HID]
    const float* sa_w2  = (const float*)d_in[8];   // [HID, 1]
    float* out = (float*)d_out;

    // -------- workspace carve-up (floats) --------
    float*    z    = (float*)d_ws;                      // NN*MM*DD = 38.4M
    float*    feat = z + (size_t)NN * MM * DD;          // NN*DD    = 12.8M
    float*    el   = feat + (size_t)NN * DD;            // NN*H
    float*    er   = el + (size_t)NN * HH;              // NN*H
    unsigned* mxu  = (unsigned*)(er + (size_t)NN * HH); // NN*H
    float*    ssum = (float*)(mxu + (size_t)NN * HH);   // NN*H
    float*    wsum = ssum + (size_t)NN * HH;            // 4
    float*    beta = wsum + 4;                          // 4

    const unsigned MONO_NEG_INF = 0x007FFFFFu;  // mono(-inf)

    // zero z accumulator and wsum every call (ws is poisoned, not re-zeroed)
    {
        size_t nz4 = (size_t)NN * MM * DD / 4;
        fill_u32x4_kernel<<<(unsigned)((nz4 + 255) / 256), 256, 0, stream>>>(
            (uint4*)z, 0u, nz4);
        fill_u32_kernel<<<1, 256, 0, stream>>>((unsigned*)wsum, 0u, 4);
    }

    for (int m = 0; m < MM; ++m) {
        const int* src = edges + (size_t)m * 2 * EE;
        const int* dst = src + EE;

        // 1) feat = h @ W[m]   (WMMA f32)
        feat_wmma_kernel<<<NN / 16, 128, 0, stream>>>(
            h, W + (size_t)m * INV * DD, feat);

        // 2) el / er
        attn_lr_kernel<<<(NN * HH + 255) / 256, 256, 0, stream>>>(
            feat, attn_l + (size_t)m * HH * FF, attn_r + (size_t)m * HH * FF,
            el, er);

        // 3) reset per-metapath accumulators
        fill_u32_kernel<<<(NN * HH + 255) / 256, 256, 0, stream>>>(
            mxu, MONO_NEG_INF, (size_t)NN * HH);
        fill_u32_kernel<<<(NN * HH + 255) / 256, 256, 0, stream>>>(
            (unsigned*)ssum, 0u, (size_t)NN * HH);

        // 4) edge softmax (max, sum) + aggregation
        edge_max_kernel<<<(EE * HH + 255) / 256, 256, 0, stream>>>(
            src, dst, el, er, mxu);
        edge_sum_kernel<<<(EE * HH + 255) / 256, 256, 0, stream>>>(
            src, dst, el, er, mxu, ssum);
        edge_agg_kernel<<<(unsigned)(((long long)EE * 32 + 255) / 256), 256, 0, stream>>>(
            src, dst, el, er, mxu, ssum, feat, z, m);
    }

    // 5) z += bias
    bias_add_kernel<<<(unsigned)(((size_t)NN * MM * DD / 4 + 255) / 256), 256, 0, stream>>>(
        (float4*)z, (const float4*)bias);

    // 6) semantic attention -> beta
    semantic_kernel<<<NN * MM, 64, 0, stream>>>(z, sa_w1, sa_b1, sa_w2, wsum);
    beta_kernel<<<1, 1, 0, stream>>>(wsum, beta);

    // 7) weighted combine
    combine_kernel<<<(NN * DD / 4 + 255) / 256, 256, 0, stream>>>(
        (const float4*)z, beta, (float4*)out);
}